// NeuralSparse_System_20916490731928
// MI455X (gfx1250) — compile-verified
//
#include <hip/hip_runtime.h>
#include <hip/hip_bf16.h>
#include <math.h>

// ---------------------------------------------------------------------------
// NeuralSparse GNN pipeline for MI455X (gfx1250, wave32).
// Dense GEMMs use V_WMMA_F32_16X16X4_F32 (native fp32 matrix path).
// Edge gather/scatter stages (the bandwidth-dominant part) use coalesced
// per-(edge,channel) threads with f32 global atomics.
// ---------------------------------------------------------------------------

typedef float v2f __attribute__((ext_vector_type(2)));
typedef float v8f __attribute__((ext_vector_type(8)));

#define NS_H    8
#define NS_C    16
#define NS_D    128
#define NS_IN   128
#define NS_SH   64
#define NS_HID4 64
#define NS_OUT  40

static inline int nblk(long long n, int b) { return (int)((n + b - 1) / b); }

// ---- float atomic max via integer atomics (correct for all finite values) --
__device__ __forceinline__ void atomicMaxF32(float* addr, float val) {
    if (val >= 0.0f) atomicMax((int*)addr, __float_as_int(val));
    else             atomicMin((unsigned int*)addr, __float_as_uint(val));
}

// ---------------------------------------------------------------------------
// Generic fill
// ---------------------------------------------------------------------------
__global__ void ns_fill(float* __restrict__ p, float v, long long n) {
    long long i = (long long)blockIdx.x * blockDim.x + threadIdx.x;
    if (i < n) p[i] = v;
}

__global__ void ns_copy(const float* __restrict__ a, float* __restrict__ b, long long n) {
    long long i = (long long)blockIdx.x * blockDim.x + threadIdx.x;
    if (i < n) b[i] = a[i];
}

// ---------------------------------------------------------------------------
// WMMA fp32 GEMM:  out[M,Nout] = X[M,K] @ W[Nout,K]^T (+bias)
//   ldx / ldw are row strides of X / W.  M%16==0, Nout%64==0? no: Nout%16==0,
//   grid = (M/16, Nout/64), block = (32,4): each wave owns one 16x16 tile.
//   16x16x4 f32 layout: lanes 0-15 hold K={0,1}, lanes 16-31 hold K={2,3};
//   D-matrix: VGPR r = row (m0 + r + 8*(lane>=16)), col n0+(lane&15).
// ---------------------------------------------------------------------------
__global__ void ns_wmma_gemm(const float* __restrict__ X, const float* __restrict__ W,
                             const float* __restrict__ bias, float* __restrict__ out,
                             int M, int K, int ldx, int ldw, int Nout) {
    const int m0   = blockIdx.x * 16;
    const int n0   = (blockIdx.y * blockDim.y + threadIdx.y) * 16;
    const int lane = threadIdx.x;
    const int r    = lane & 15;
    const int kk   = (lane >> 4) << 1;           // 0 or 2

    const float* pA = X + (size_t)(m0 + r) * ldx + kk;
    const float* pB = W + (size_t)(n0 + r) * ldw + kk;

    v8f c = {0.f, 0.f, 0.f, 0.f, 0.f, 0.f, 0.f, 0.f};
    for (int k = 0; k < K; k += 4) {
        v2f a, b;
        a.x = pA[k]; a.y = pA[k + 1];
        b.x = pB[k]; b.y = pB[k + 1];
        c = __builtin_amdgcn_wmma_f32_16x16x4_f32(false, a, false, b,
                                                  (short)0, c, false, false);
    }
    const int col     = n0 + r;
    const float bv    = bias ? bias[col] : 0.0f;
    const int rowBase = m0 + ((lane >> 4) << 3);
#pragma unroll
    for (int i = 0; i < 8; ++i)
        out[(size_t)(rowBase + i) * Nout + col] = c[i] + bv;
}

// ---------------------------------------------------------------------------
// GAT attention coefficients: out[n,h] = sum_c h[n,h*16+c] * att[h,c]
// ---------------------------------------------------------------------------
__global__ void ns_attn_coeff(const float* __restrict__ h, const float* __restrict__ att,
                              float* __restrict__ out, int N) {
    int i = blockIdx.x * blockDim.x + threadIdx.x;
    if (i >= N * NS_H) return;
    int n = i >> 3, hd = i & 7;
    const float* hp = h + (size_t)n * NS_D + hd * NS_C;
    const float* ap = att + hd * NS_C;
    float s = 0.f;
#pragma unroll
    for (int c = 0; c < NS_C; ++c) s += hp[c] * ap[c];
    out[i] = s;
}

// ---------------------------------------------------------------------------
// GAT edge passes (E real edges + N implicit self-loops)
// ---------------------------------------------------------------------------
__global__ void ns_edge_max(const int* __restrict__ row, const int* __restrict__ col,
                            int E, int N, const float* __restrict__ a_src,
                            const float* __restrict__ a_dst,
                            float* __restrict__ ebuf, float* __restrict__ m) {
    long long i = (long long)blockIdx.x * blockDim.x + threadIdx.x;
    long long tot = (long long)(E + N) * NS_H;
    if (i >= tot) return;
    int e = (int)(i >> 3), hd = (int)(i & 7);
    int s, d;
    if (e < E) { s = row[e]; d = col[e]; } else { s = d = e - E; }
    float v = a_src[s * NS_H + hd] + a_dst[d * NS_H + hd];
    v = v > 0.f ? v : 0.2f * v;                 // leaky_relu(0.2)
    ebuf[i] = v;
    atomicMaxF32(&m[d * NS_H + hd], v);
}

__global__ void ns_edge_expsum(const int* __restrict__ row, const int* __restrict__ col,
                               int E, int N, const float* __restrict__ m,
                               float* __restrict__ ebuf, float* __restrict__ denom) {
    long long i = (long long)blockIdx.x * blockDim.x + threadIdx.x;
    long long tot = (long long)(E + N) * NS_H;
    if (i >= tot) return;
    int e = (int)(i >> 3), hd = (int)(i & 7);
    int d = (e < E) ? col[e] : (e - E);
    float v = __expf(ebuf[i] - m[d * NS_H + hd]);
    ebuf[i] = v;
    atomicAdd(&denom[d * NS_H + hd], v);
}

__global__ void ns_edge_msg(const int* __restrict__ row, const int* __restrict__ col,
                            int E, int N, const float* __restrict__ h,
                            const float* __restrict__ ebuf, const float* __restrict__ denom,
                            float* __restrict__ out) {
    long long i = (long long)blockIdx.x * blockDim.x + threadIdx.x;
    long long tot = (long long)(E + N) * NS_D;
    if (i >= tot) return;
    int e = (int)(i >> 7), dch = (int)(i & 127), hd = dch >> 4;
    int s, d;
    if (e < E) { s = row[e]; d = col[e]; } else { s = d = e - E; }
    float alpha = ebuf[(size_t)e * NS_H + hd] / (denom[d * NS_H + hd] + 1e-16f);
    atomicAdd(&out[(size_t)d * NS_D + dch], h[(size_t)s * NS_D + dch] * alpha);
}

// ---------------------------------------------------------------------------
// BatchNorm (over rows) : stats = [sum | sumsq | mean | rsqrt(var+eps)]
// ---------------------------------------------------------------------------
__global__ void ns_bn_stats(const float* __restrict__ x, float* __restrict__ sums,
                            int Nrows, int D) {
    int c = threadIdx.x;                         // blockDim.x == D
    float s = 0.f, s2 = 0.f;
    for (int rw = blockIdx.x; rw < Nrows; rw += gridDim.x) {
        float v = x[(size_t)rw * D + c];
        s += v; s2 += v * v;
    }
    atomicAdd(&sums[c], s);
    atomicAdd(&sums[D + c], s2);
}

__global__ void ns_bn_finalize(float* __restrict__ stats, int Nrows, int D) {
    int c = threadIdx.x;
    if (c >= D) return;
    float mu  = stats[c] / (float)Nrows;
    float var = stats[D + c] / (float)Nrows - mu * mu;
    stats[2 * D + c] = mu;
    stats[3 * D + c] = rsqrtf(var + 1e-5f);
}

// act: 0 = ELU, 1 = ReLU.  mask = D-1 (D power of two).
__global__ void ns_bn_apply(const float* __restrict__ x, const float* __restrict__ stats,
                            const float* __restrict__ gamma, const float* __restrict__ beta,
                            float* __restrict__ out, long long total, int mask, int D, int act) {
    long long i = (long long)blockIdx.x * blockDim.x + threadIdx.x;
    if (i >= total) return;
    int c = (int)(i & mask);
    float v = (x[i] - stats[2 * D + c]) * stats[3 * D + c] * gamma[c] + beta[c];
    out[i] = act ? fmaxf(v, 0.f) : (v > 0.f ? v : __expf(v) - 1.f);
}

// ---------------------------------------------------------------------------
// Edge scorer + straight-through Gumbel:  weights[e] = 1[lr + g1 > g0]
// ---------------------------------------------------------------------------
__global__ void ns_edge_score(const int* __restrict__ row, const int* __restrict__ col, int E,
                              const float* __restrict__ pr, const float* __restrict__ pc,
                              const float* __restrict__ bs1, const float* __restrict__ Ws2,
                              const float* __restrict__ bs2, const float* __restrict__ gum,
                              float* __restrict__ out_w, float* __restrict__ out_lr) {
    int e = blockIdx.x * blockDim.x + threadIdx.x;
    if (e >= E) return;
    const float* a = pr + (size_t)row[e] * NS_SH;
    const float* b = pc + (size_t)col[e] * NS_SH;
    float acc = 0.f;
#pragma unroll 8
    for (int c = 0; c < NS_SH; ++c) {
        float v = a[c] + b[c] + bs1[c];
        acc += fmaxf(v, 0.f) * Ws2[c];
    }
    float lr = acc + bs2[0];
    out_lr[e] = lr;
    out_w[e]  = (lr + gum[2 * e + 1] > gum[2 * e]) ? 1.0f : 0.0f;
}

// agg[row[e]] += w[e] * h_base[col[e]]   (into hs, pre-seeded with h_base)
__global__ void ns_sparse_agg(const int* __restrict__ row, const int* __restrict__ col, int E,
                              const float* __restrict__ w, const float* __restrict__ hb,
                              float* __restrict__ hs) {
    long long i = (long long)blockIdx.x * blockDim.x + threadIdx.x;
    if (i >= (long long)E * NS_D) return;
    int e = (int)(i >> 7), dch = (int)(i & 127);
    float we = w[e];
    if (we != 0.0f)
        atomicAdd(&hs[(size_t)row[e] * NS_D + dch], we * hb[(size_t)col[e] * NS_D + dch]);
}

// ---------------------------------------------------------------------------
// Final classifier (OUT=40 not a 16-multiple -> VALU GEMV) + log_softmax
// ---------------------------------------------------------------------------
__global__ void ns_cls_logits(const float* __restrict__ cb, const float* __restrict__ Wc2,
                              const float* __restrict__ bc2, float* __restrict__ lg, int N) {
    int i = blockIdx.x * blockDim.x + threadIdx.x;
    if (i >= N * NS_OUT) return;
    int n = i / NS_OUT, o = i - n * NS_OUT;
    const float* cp = cb + (size_t)n * NS_HID4;
    const float* wp = Wc2 + o * NS_HID4;
    float acc = bc2[o];
#pragma unroll 8
    for (int k = 0; k < NS_HID4; ++k) acc += cp[k] * wp[k];
    lg[i] = acc;
}

__global__ void ns_log_softmax(const float* __restrict__ lg, float* __restrict__ out, int N) {
    int n = blockIdx.x * blockDim.x + threadIdx.x;
    if (n >= N) return;
    const float* p = lg + (size_t)n * NS_OUT;
    float mx = p[0];
    for (int o = 1; o < NS_OUT; ++o) mx = fmaxf(mx, p[o]);
    float s = 0.f;
    for (int o = 0; o < NS_OUT; ++o) s += __expf(p[o] - mx);
    float ls = __logf(s);
    for (int o = 0; o < NS_OUT; ++o) out[(size_t)n * NS_OUT + o] = p[o] - mx - ls;
}

// ---------------------------------------------------------------------------
// Host launcher
// ---------------------------------------------------------------------------
extern "C" void kernel_launch(void* const* d_in, const int* in_sizes, int n_in,
                              void* d_out, int out_size, void* d_ws, size_t ws_size,
                              hipStream_t stream) {
    const float* x     = (const float*)d_in[0];
    const int*   ei    = (const int*)d_in[1];
    const float* gum   = (const float*)d_in[2];
    const float* W_res = (const float*)d_in[3];
    const float* b_res = (const float*)d_in[4];
    const float* Wg[2] = {(const float*)d_in[5],  (const float*)d_in[11]};
    const float* as_[2]= {(const float*)d_in[6],  (const float*)d_in[12]};
    const float* ad_[2]= {(const float*)d_in[7],  (const float*)d_in[13]};
    // bg1/bg2 (d_in[8], d_in[14]) cancel exactly under batchnorm -> skipped
    const float* gg[2] = {(const float*)d_in[9],  (const float*)d_in[15]};
    const float* bb[2] = {(const float*)d_in[10], (const float*)d_in[16]};
    const float* Ws1 = (const float*)d_in[17];
    const float* bs1 = (const float*)d_in[18];
    const float* Ws2 = (const float*)d_in[19];
    const float* bs2 = (const float*)d_in[20];
    const float* Wc1 = (const float*)d_in[21];
    const float* bc1 = (const float*)d_in[22];
    const float* gc  = (const float*)d_in[23];
    const float* bec = (const float*)d_in[24];
    const float* Wc2 = (const float*)d_in[25];
    const float* bc2 = (const float*)d_in[26];

    const int N = in_sizes[0] / NS_IN;      // 50000 (multiple of 16)
    const int E = in_sizes[1] / 2;          // 800000
    const int* row = ei;
    const int* col = ei + E;
    const long long ND = (long long)N * NS_D;

    // ---- workspace layout (floats) ----
    float* W0 = (float*)d_ws;
    size_t off = 0;
    auto alloc = [&](size_t n) { float* p = W0 + off; off += n; return p; };
    float* bufA  = alloc(ND);                       // x_proj -> h1 -> h_base
    float* bufB  = alloc(ND);                       // W-projected feats -> h_sparse
    float* bufC  = alloc(ND);                       // GAT aggregation
    float* a_src = alloc((size_t)N * NS_H);
    float* a_dst = alloc((size_t)N * NS_H);
    float* mbuf  = alloc((size_t)N * NS_H);
    float* dbuf  = alloc((size_t)N * NS_H);
    float* ebuf  = alloc((size_t)(E + N) * NS_H);
    float* pr    = alloc((size_t)N * NS_SH);
    float* pc    = alloc((size_t)N * NS_SH);
    float* cbuf  = alloc((size_t)N * NS_HID4);
    float* lbuf  = alloc((size_t)N * NS_OUT);
    float* stats = alloc(4 * NS_D);
    if (off * sizeof(float) > ws_size) return;      // defensive; deterministic

    float* out_lsm = (float*)d_out;                 // N*OUT
    float* out_w   = out_lsm + (size_t)N * NS_OUT;  // E
    float* out_lr  = out_w + E;                     // E

    const long long eh_tot = (long long)(E + N) * NS_H;
    const long long ed_tot = (long long)(E + N) * NS_D;
    dim3 gblk(32, 4);

    // 1) x_proj = x @ W_res^T + b_res
    ns_wmma_gemm<<<dim3(N / 16, NS_D / 64), gblk, 0, stream>>>(
        x, W_res, b_res, bufA, N, NS_IN, NS_IN, NS_IN, NS_D);

    // 2) two GAT layers: bufA -> bufA
    for (int L = 0; L < 2; ++L) {
        ns_wmma_gemm<<<dim3(N / 16, NS_D / 64), gblk, 0, stream>>>(
            bufA, Wg[L], nullptr, bufB, N, NS_D, NS_D, NS_D, NS_D);
        ns_attn_coeff<<<nblk((long long)N * NS_H, 256), 256, 0, stream>>>(bufB, as_[L], a_src, N);
        ns_attn_coeff<<<nblk((long long)N * NS_H, 256), 256, 0, stream>>>(bufB, ad_[L], a_dst, N);
        ns_fill<<<nblk((long long)N * NS_H, 256), 256, 0, stream>>>(mbuf, -3.0e38f, (long long)N * NS_H);
        ns_fill<<<nblk((long long)N * NS_H, 256), 256, 0, stream>>>(dbuf, 0.f, (long long)N * NS_H);
        ns_fill<<<nblk(ND, 256), 256, 0, stream>>>(bufC, 0.f, ND);
        ns_edge_max<<<nblk(eh_tot, 256), 256, 0, stream>>>(row, col, E, N, a_src, a_dst, ebuf, mbuf);
        ns_edge_expsum<<<nblk(eh_tot, 256), 256, 0, stream>>>(row, col, E, N, mbuf, ebuf, dbuf);
        ns_edge_msg<<<nblk(ed_tot, 256), 256, 0, stream>>>(row, col, E, N, bufB, ebuf, dbuf, bufC);
        ns_fill<<<1, 2 * NS_D, 0, stream>>>(stats, 0.f, 2 * NS_D);
        ns_bn_stats<<<256, NS_D, 0, stream>>>(bufC, stats, N, NS_D);
        ns_bn_finalize<<<1, NS_D, 0, stream>>>(stats, N, NS_D);
        ns_bn_apply<<<nblk(ND, 256), 256, 0, stream>>>(
            bufC, stats, gg[L], bb[L], bufA, ND, NS_D - 1, NS_D, /*ELU*/0);
    }
    // bufA == h_base

    // 3) edge scorer inputs: pr = h@Ws1[:, :D]^T, pc = h@Ws1[:, D:]^T  (ldw=256)
    ns_wmma_gemm<<<dim3(N / 16, 1), gblk, 0, stream>>>(
        bufA, Ws1,          nullptr, pr, N, NS_D, NS_D, 2 * NS_D, NS_SH);
    ns_wmma_gemm<<<dim3(N / 16, 1), gblk, 0, stream>>>(
        bufA, Ws1 + NS_D,   nullptr, pc, N, NS_D, NS_D, 2 * NS_D, NS_SH);

    // 4) per-edge MLP + straight-through Gumbel -> weights / logits_raw
    ns_edge_score<<<nblk(E, 256), 256, 0, stream>>>(
        row, col, E, pr, pc, bs1, Ws2, bs2, gum, out_w, out_lr);

    // 5) h_sparse = h_base + segment_sum(w * h_base[col] -> row)
    ns_copy<<<nblk(ND, 256), 256, 0, stream>>>(bufA, bufB, ND);
    ns_sparse_agg<<<nblk((long long)E * NS_D, 256), 256, 0, stream>>>(
        row, col, E, out_w, bufA, bufB);

    // 6) classifier: c = relu(bn(h_sparse @ Wc1^T + bc1))
    ns_wmma_gemm<<<dim3(N / 16, 1), gblk, 0, stream>>>(
        bufB, Wc1, bc1, cbuf, N, NS_D, NS_D, NS_D, NS_HID4);
    ns_fill<<<1, 2 * NS_HID4, 0, stream>>>(stats, 0.f, 2 * NS_HID4);
    ns_bn_stats<<<256, NS_HID4, 0, stream>>>(cbuf, stats, N, NS_HID4);
    ns_bn_finalize<<<1, NS_HID4, 0, stream>>>(stats, N, NS_HID4);
    ns_bn_apply<<<nblk((long long)N * NS_HID4, 256), 256, 0, stream>>>(
        cbuf, stats, gc, bec, cbuf, (long long)N * NS_HID4, NS_HID4 - 1, NS_HID4, /*ReLU*/1);

    // 7) logits + log_softmax
    ns_cls_logits<<<nblk((long long)N * NS_OUT, 256), 256, 0, stream>>>(cbuf, Wc2, bc2, lbuf, N);
    ns_log_softmax<<<nblk(N, 256), 256, 0, stream>>>(lbuf, out_lsm, N);
}